// MINDSSC_1005022347626
// MI455X (gfx1250) — compile-verified
//
#include <hip/hip_runtime.h>
#include <hip/hip_bf16.h>

// ---------------------------------------------------------------------------
// MIND-SSC loss for two [1,1,128,128,128] fp32 volumes on gfx1250 (MI455X).
//
// Pipeline (all on `stream`, graph-capture safe, deterministic):
//   pass1   : per-voxel mind_var, block partial sums        -> ws
//   reduce1 : global mean of mind_var per volume            -> ws header
//   pass2   : recompute SSD, clamp, exp, MSE partial sums   -> ws
//   reduce2 : final mean                                    -> d_out[0]
//
// CDNA5 usage:
//  - TENSOR_LOAD_TO_LDS (TDM): stages the per-block img tile into LDS once;
//    all stencil field reads become LDS reads (tile reused ~30x/element).
//  - s_wait_tensorcnt for TDM completion.
//  - V_WMMA_F32_16X16X4_F32: all-ones A => layout-independent exact f32 sum.
//  - ds_swizzle_b32 xor folds for the wave32 cross-lane reduction.
// ---------------------------------------------------------------------------

#define NVOX (128 * 128 * 128)

typedef float v2f __attribute__((ext_vector_type(2)));
typedef float v8f __attribute__((ext_vector_type(8)));
typedef unsigned int v4u __attribute__((ext_vector_type(4)));
typedef int v4i __attribute__((ext_vector_type(4)));
typedef int v8i __attribute__((ext_vector_type(8)));
typedef unsigned int u32;

// Six shifted fields, in the reference's `six` order:
//   0:H-2  1:D-2  2:W-2  3:D+2  4:H+2  5:W+2
__constant__ int F_DH[6] = {-2, 0,  0, 0, 2, 0};
__constant__ int F_DW[6] = { 0, 0, -2, 0, 0, 2};
__constant__ int F_DD[6] = { 0,-2,  0, 2, 0, 0};
// 12 channel pairs (i>j, squared-distance 2), row-major nonzero order:
__constant__ int C_F1[12] = {1, 2, 2, 3, 3, 4, 4, 4, 5, 5, 5, 5};
__constant__ int C_F2[12] = {0, 0, 1, 0, 2, 1, 2, 3, 0, 1, 3, 4};

__device__ __forceinline__ int imaxi(int a, int b) { return a > b ? a : b; }
__device__ __forceinline__ int imini(int a, int b) { return a < b ? a : b; }
__device__ __forceinline__ int clampi(int x) { return imini(127, imaxi(0, x)); }

// ---------------------------------------------------------------------------
// TDM: DMA a 3-D img tile [nh rows][nw cols][128 d] (corner gh0,gw0, element
// strides 16384/128/1 in f32 units) into LDS at lds_off, packed as
// [ih][iw][d] (pitch nw*128). Descriptor per CDNA5 ISA ch.8 (D# groups 0-3).
// Issued by one wave; EXEC is ignored by tensor ops.
// ---------------------------------------------------------------------------
__device__ __forceinline__ void tdm_load_tile(const float* __restrict__ img,
                                              u32 lds_off, int gh0, int nh,
                                              int gw0, int nw) {
  const unsigned long long gaddr =
      (unsigned long long)(uintptr_t)(img + ((size_t)gh0 << 14) + ((size_t)gw0 << 7));
  v4u g0;
  g0[0] = 1u;                                     // count=1 (valid descriptor)
  g0[1] = lds_off;                                // lds_addr (bytes)
  g0[2] = (u32)gaddr;                             // global_addr[31:0]
  g0[3] = (u32)((gaddr >> 32) & 0x01FFFFFFull) |  // global_addr[56:32]
          (2u << 30);                             // type=2 ("image")
  v8i g1;
  g1[0] = (int)(2u << 16);                        // data_size=2 (4 bytes)
  g1[1] = (int)(128u << 16);                      // tensor_dim0[15:0]=128
  g1[2] = (int)((u32)nw << 16);                   // tensor_dim1[15:0]=nw
  g1[3] = (int)(128u << 16);                      // tile_dim0=128
  g1[4] = (int)((u32)nw | ((u32)nh << 16));       // tile_dim1=nw, tile_dim2=nh
  g1[5] = 128;                                    // tensor_dim0_stride=128
  g1[6] = (int)(16384u << 16);                    // tensor_dim1_stride lo=16384
  g1[7] = 0;                                      // tensor_dim1_stride hi
  v4i g2;
  g2[0] = nh;                                     // tensor_dim2=nh
  g2[1] = 0; g2[2] = 0; g2[3] = 0;                // no dim3 / iterate
  v4i g3; g3[0] = 0; g3[1] = 0; g3[2] = 0; g3[3] = 0;
#if __clang_major__ >= 23
  v8i g4; g4[0]=0; g4[1]=0; g4[2]=0; g4[3]=0; g4[4]=0; g4[5]=0; g4[6]=0; g4[7]=0;
  __builtin_amdgcn_tensor_load_to_lds(g0, g1, g2, g3, g4, 0);
#else
  __builtin_amdgcn_tensor_load_to_lds(g0, g1, g2, g3, 0);
#endif
}

// Field read from the staged LDS tile. (h,w) is an already-clamped box
// position; the +-2 field shift is clamped again (nested-clamp semantics),
// and every resulting coordinate provably lies inside the staged tile.
__device__ __forceinline__ float loadFieldT(const float* __restrict__ tile,
                                            int gh0, int gw0, int nw,
                                            int h, int w, int d, int f) {
  const int ih = clampi(h + F_DH[f]) - gh0;
  const int iw = clampi(w + F_DW[f]) - gw0;
  const int id = clampi(d + F_DD[f]);
  return tile[(ih * nw + iw) * 128 + id];
}

// ---------------------------------------------------------------------------
// Separable 5x5x5 box-filtered squared-difference ("raw SSD", unnormalized;
// caller multiplies by 1/125) for one channel c over the block tile
// (h0, w0..w0+WT-1, d=threadIdx.x).  blockDim.x == 128 (one thread per d).
// ---------------------------------------------------------------------------
template <int WT>
__device__ __forceinline__ void ssd_channel(const float* __restrict__ tile,
                                            int gh0, int gw0, int nw,
                                            int h0, int w0, int c,
                                            float* __restrict__ diffS,
                                            float* __restrict__ As,
                                            float* __restrict__ acc) {
  constexpr int WP = WT + 4;
  const int d = threadIdx.x;
  const int f1 = C_F1[c];
  const int f2 = C_F2[c];
  for (int dh = 0; dh < 5; ++dh) {
    const int hh = clampi(h0 + dh - 2);
    // Stage 1: squared diff at (hh, w', d) for the W halo range (LDS tile reads).
    #pragma unroll
    for (int iw = 0; iw < WP; ++iw) {
      const int ww = clampi(w0 - 2 + iw);
      const float a = loadFieldT(tile, gh0, gw0, nw, hh, ww, d, f1);
      const float b = loadFieldT(tile, gh0, gw0, nw, hh, ww, d, f2);
      const float t = a - b;
      diffS[iw * 128 + d] = t * t;
    }
    __syncthreads();
    // Stage 2: edge-clamped 5-tap sum along D (contiguous axis) via LDS.
    #pragma unroll
    for (int iw = 0; iw < WP; ++iw) {
      const float* row = diffS + iw * 128;
      As[iw * 128 + d] = row[imaxi(d - 2, 0)] + row[imaxi(d - 1, 0)] + row[d] +
                         row[imini(d + 1, 127)] + row[imini(d + 2, 127)];
    }
    __syncthreads();
    // Stage 3: 5-tap sum along W into per-output accumulators (H via dh loop).
    #pragma unroll
    for (int jw = 0; jw < WT; ++jw) {
      acc[jw] += As[(jw + 0) * 128 + d] + As[(jw + 1) * 128 + d] +
                 As[(jw + 2) * 128 + d] + As[(jw + 3) * 128 + d] +
                 As[(jw + 4) * 128 + d];
    }
    __syncthreads();
  }
}

// ---------------------------------------------------------------------------
// Block-wide exact f32 sum using V_WMMA_F32_16X16X4_F32.
// With A = all-ones (16x4), D[m,n] = sum_k B[k,n], so sum over all 256 D
// elements = 16 * sum(B) regardless of the VGPR<->matrix layout. Cross-lane
// folds use pure-xor ds_swizzle (wave32), then a tiny cross-wave LDS fold.
// Called uniformly by all 128 threads (EXEC all ones, as WMMA requires).
// ---------------------------------------------------------------------------
__device__ __forceinline__ float block_sum_wmma(float v, float* __restrict__ red) {
  v2f a; a[0] = 1.0f; a[1] = 1.0f;
  v2f b; b[0] = v;    b[1] = 0.0f;
  v8f cz = {0.f, 0.f, 0.f, 0.f, 0.f, 0.f, 0.f, 0.f};
  v8f dm = __builtin_amdgcn_wmma_f32_16x16x4_f32(false, a, false, b,
                                                 (short)0, cz, false, false);
  float s = dm[0] + dm[1] + dm[2] + dm[3] + dm[4] + dm[5] + dm[6] + dm[7];
  // xor-swizzle folds: imm = (xor_mask << 10) | 0x1f (and=0x1f, or=0)
  s += __int_as_float(__builtin_amdgcn_ds_swizzle(__float_as_int(s), 0x401F)); // ^16
  s += __int_as_float(__builtin_amdgcn_ds_swizzle(__float_as_int(s), 0x201F)); // ^8
  s += __int_as_float(__builtin_amdgcn_ds_swizzle(__float_as_int(s), 0x101F)); // ^4
  s += __int_as_float(__builtin_amdgcn_ds_swizzle(__float_as_int(s), 0x081F)); // ^2
  s += __int_as_float(__builtin_amdgcn_ds_swizzle(__float_as_int(s), 0x041F)); // ^1
  s *= 0.0625f;  // /16: each column-sum of B appears in 16 rows of D
  const int lane = threadIdx.x & 31;
  const int wv = threadIdx.x >> 5;
  if (lane == 0) red[wv] = s;
  __syncthreads();
  const float tot = red[0] + red[1] + red[2] + red[3];
  __syncthreads();
  return tot;
}

// Dynamic-LDS float counts (320 KB/WGP on gfx1250; 2 blocks/WGP resident).
#define TILE1_F (9 * 16 * 128) /* pass1: 73.7 KB */
#define TILE2_F (9 * 12 * 128) /* pass2: 55.3 KB */

// ---------------------------------------------------------------------------
// Pass 1: per-voxel mind_var = mean_c(ssd) - min_c(ssd), scaled by 1/125,
// summed per block. grid = (16 w-tiles, 128 h, 2 volumes), block = 128.
// ---------------------------------------------------------------------------
__global__ __launch_bounds__(128) void mind_pass1(const float* __restrict__ img0,
                                                  const float* __restrict__ img1,
                                                  float* __restrict__ partials) {
  extern __shared__ float smem[];
  float* tile  = smem;                    // [nh][nw][128]
  float* diffS = smem + TILE1_F;          // 12*128
  float* As    = diffS + 12 * 128;        // 12*128
  float* red   = As + 12 * 128;           // 4 (+pad)
  const int wtile = blockIdx.x;
  const int h0 = blockIdx.y;
  const int vol = blockIdx.z;
  const float* img = vol ? img1 : img0;
  const int w0 = wtile * 8;

  const int gh0 = imaxi(h0 - 4, 0), nh = imini(h0 + 4, 127) - gh0 + 1;
  const int gw0 = imaxi(w0 - 4, 0), nw = imini(w0 + 11, 127) - gw0 + 1;
  if (threadIdx.x < 32) {
    tdm_load_tile(img, (u32)(uintptr_t)tile, gh0, nh, gw0, nw);
    __builtin_amdgcn_s_wait_tensorcnt(0);
  }
  __syncthreads();

  float minv[8], sumv[8];
  #pragma unroll
  for (int j = 0; j < 8; ++j) { minv[j] = 3.4e38f; sumv[j] = 0.0f; }

  for (int c = 0; c < 12; ++c) {
    float acc[8];
    #pragma unroll
    for (int j = 0; j < 8; ++j) acc[j] = 0.0f;
    ssd_channel<8>(tile, gh0, gw0, nw, h0, w0, c, diffS, As, acc);
    #pragma unroll
    for (int j = 0; j < 8; ++j) {
      minv[j] = fminf(minv[j], acc[j]);
      sumv[j] += acc[j];
    }
  }
  float local = 0.0f;
  #pragma unroll
  for (int j = 0; j < 8; ++j)
    local += (sumv[j] * (1.0f / 12.0f) - minv[j]) * (1.0f / 125.0f);

  const float tot = block_sum_wmma(local, red);
  if (threadIdx.x == 0) partials[(vol * 128 + h0) * 16 + wtile] = tot;
}

// ---------------------------------------------------------------------------
// Pass 2: recompute SSD for both volumes, clamp var with the global means,
// descriptors exp(-mind/var), accumulate squared descriptor difference.
// grid = (32 w-tiles, 128 h), block = 128.
// ---------------------------------------------------------------------------
__global__ __launch_bounds__(128) void mind_pass2(const float* __restrict__ imgP,
                                                  const float* __restrict__ imgT,
                                                  const float* __restrict__ mhdr,
                                                  float* __restrict__ partials) {
  extern __shared__ float smem[];
  float* tile  = smem;                    // [nh][nw][128]
  float* diffS = smem + TILE2_F;          // 8*128
  float* As    = diffS + 8 * 128;         // 8*128
  float* ssdP  = As + 8 * 128;            // 12*4*128, thread-private slots
  float* ssdT  = ssdP + 12 * 4 * 128;     // 12*4*128
  float* red   = ssdT + 12 * 4 * 128;     // 4 (+pad)
  const int wtile = blockIdx.x;
  const int h0 = blockIdx.y;
  const int w0 = wtile * 4;
  const int d = threadIdx.x;

  const int gh0 = imaxi(h0 - 4, 0), nh = imini(h0 + 4, 127) - gh0 + 1;
  const int gw0 = imaxi(w0 - 4, 0), nw = imini(w0 + 7, 127) - gw0 + 1;

  float minP[4], sumP[4], minT[4], sumT[4];
  #pragma unroll
  for (int j = 0; j < 4; ++j) {
    minP[j] = 3.4e38f; sumP[j] = 0.0f;
    minT[j] = 3.4e38f; sumT[j] = 0.0f;
  }

  // Volume P (y_pred)
  if (threadIdx.x < 32) {
    tdm_load_tile(imgP, (u32)(uintptr_t)tile, gh0, nh, gw0, nw);
    __builtin_amdgcn_s_wait_tensorcnt(0);
  }
  __syncthreads();
  for (int c = 0; c < 12; ++c) {
    float acc[4];
    #pragma unroll
    for (int j = 0; j < 4; ++j) acc[j] = 0.0f;
    ssd_channel<4>(tile, gh0, gw0, nw, h0, w0, c, diffS, As, acc);
    #pragma unroll
    for (int j = 0; j < 4; ++j) {
      ssdP[(c * 4 + j) * 128 + d] = acc[j];
      minP[j] = fminf(minP[j], acc[j]);
      sumP[j] += acc[j];
    }
  }
  __syncthreads();  // all tile reads done before the next TDM overwrites it

  // Volume T (y_true)
  if (threadIdx.x < 32) {
    tdm_load_tile(imgT, (u32)(uintptr_t)tile, gh0, nh, gw0, nw);
    __builtin_amdgcn_s_wait_tensorcnt(0);
  }
  __syncthreads();
  for (int c = 0; c < 12; ++c) {
    float acc[4];
    #pragma unroll
    for (int j = 0; j < 4; ++j) acc[j] = 0.0f;
    ssd_channel<4>(tile, gh0, gw0, nw, h0, w0, c, diffS, As, acc);
    #pragma unroll
    for (int j = 0; j < 4; ++j) {
      ssdT[(c * 4 + j) * 128 + d] = acc[j];
      minT[j] = fminf(minT[j], acc[j]);
      sumT[j] += acc[j];
    }
  }

  const float mP = mhdr[0];
  const float mT = mhdr[1];
  float varP[4], varT[4];
  #pragma unroll
  for (int j = 0; j < 4; ++j) {
    const float vp = (sumP[j] * (1.0f / 12.0f) - minP[j]) * (1.0f / 125.0f);
    const float vt = (sumT[j] * (1.0f / 12.0f) - minT[j]) * (1.0f / 125.0f);
    varP[j] = fminf(fmaxf(vp, mP * 0.001f), mP * 1000.0f);
    varT[j] = fminf(fmaxf(vt, mT * 0.001f), mT * 1000.0f);
  }

  float local = 0.0f;
  for (int c = 0; c < 12; ++c) {
    #pragma unroll
    for (int j = 0; j < 4; ++j) {
      const float mp = (ssdP[(c * 4 + j) * 128 + d] - minP[j]) * (1.0f / 125.0f);
      const float mt = (ssdT[(c * 4 + j) * 128 + d] - minT[j]) * (1.0f / 125.0f);
      const float ep = expf(-mp / varP[j]);
      const float et = expf(-mt / varT[j]);
      const float t = et - ep;
      local += t * t;
    }
  }

  const float tot = block_sum_wmma(local, red);
  if (threadIdx.x == 0) partials[h0 * 32 + wtile] = tot;
}

// ---------------------------------------------------------------------------
// Deterministic single-block reduction: out[v] = scale * sum(in[v*n .. +n)).
// ---------------------------------------------------------------------------
__global__ __launch_bounds__(256) void reduce_scaled(const float* __restrict__ in,
                                                     float* __restrict__ out,
                                                     int n, int nout, float scale) {
  __shared__ float sdata[256];
  for (int v = 0; v < nout; ++v) {
    float s = 0.0f;
    for (int i = threadIdx.x; i < n; i += 256) s += in[v * n + i];
    sdata[threadIdx.x] = s;
    __syncthreads();
    for (int stride = 128; stride > 0; stride >>= 1) {
      if (threadIdx.x < (unsigned)stride) sdata[threadIdx.x] += sdata[threadIdx.x + stride];
      __syncthreads();
    }
    if (threadIdx.x == 0) out[v] = sdata[0] * scale;
    __syncthreads();
  }
}

// ---------------------------------------------------------------------------
extern "C" void kernel_launch(void* const* d_in, const int* in_sizes, int n_in,
                              void* d_out, int out_size, void* d_ws, size_t ws_size,
                              hipStream_t stream) {
  const float* pred = (const float*)d_in[0];  // y_pred
  const float* truth = (const float*)d_in[1]; // y_true
  float* ws = (float*)d_ws;
  float* mhdr = ws;            // [0]=mean(var_pred), [1]=mean(var_true)
  float* p1 = ws + 16;         // 2 volumes x 2048 block partials
  float* p2 = ws + 16 + 4096;  // 4096 block partials

  const size_t sh1 = (TILE1_F + 12 * 128 + 12 * 128 + 8) * sizeof(float);
  const size_t sh2 =
      (TILE2_F + 8 * 128 + 8 * 128 + 12 * 4 * 128 + 12 * 4 * 128 + 8) * sizeof(float);

  dim3 blk(128, 1, 1);
  dim3 g1(16, 128, 2);
  mind_pass1<<<g1, blk, sh1, stream>>>(pred, truth, p1);
  reduce_scaled<<<1, 256, 0, stream>>>(p1, mhdr, 2048, 2, 1.0f / (float)NVOX);
  dim3 g2(32, 128, 1);
  mind_pass2<<<g2, blk, sh2, stream>>>(pred, truth, mhdr, p2);
  reduce_scaled<<<1, 256, 0, stream>>>(p2, (float*)d_out, 4096, 1,
                                       1.0f / (12.0f * (float)NVOX));
}